// GATConv_77953656422437
// MI455X (gfx1250) — compile-verified
//
#include <hip/hip_runtime.h>
#include <math.h>
#include <stdint.h>

#define NN 50000
#define NE 1600000
#define IN_CH 128
#define HEADS 4
#define OUT_CH 32
#define HD 128          // HEADS*OUT_CH
#define NEG_SLOPE 0.2f

typedef __attribute__((ext_vector_type(2))) float v2f;
typedef __attribute__((ext_vector_type(8))) float v8f;

// order-preserving float <-> uint mapping for exact atomic max
__device__ __forceinline__ unsigned f2mono(float f) {
    unsigned u = __float_as_uint(f);
    return (u & 0x80000000u) ? ~u : (u | 0x80000000u);
}
__device__ __forceinline__ float mono2f(unsigned u) {
    return __uint_as_float((u & 0x80000000u) ? (u & 0x7FFFFFFFu) : ~u);
}

// ---------------------------------------------------------------------------
// Kernel 1: h = x @ W via V_WMMA_F32_16X16X4_F32.
// W (64 KB) is staged into LDS once per block with GLOBAL_LOAD_ASYNC_TO_LDS_B128
// (ASYNCcnt path), then B fragments come from ds_load (low latency) while A
// rows stream from global. One wave owns a 16-row x 128-col tile.
// ---------------------------------------------------------------------------
__global__ __launch_bounds__(256) void gat_gemm_wmma(
    const float* __restrict__ x, const float* __restrict__ w,
    float* __restrict__ h)
{
    __shared__ float sw[IN_CH * HD];            // 64 KB: w[k][n]

    // cooperative async fill: 4096 x 16B chunks, 16 per thread
    {
        const int t = threadIdx.x;
        const unsigned lbase = (unsigned)(uintptr_t)(&sw[0]);
#pragma unroll
        for (int i = 0; i < 16; ++i) {
            const int chunk = t + i * 256;                  // 16-byte chunk id
            const unsigned loff = lbase + chunk * 16;       // LDS byte offset
            const float* gp = w + chunk * 4;                // global source
            asm volatile("global_load_async_to_lds_b128 %0, %1, off"
                         :: "v"(loff), "v"(gp) : "memory");
        }
        asm volatile("s_wait_asynccnt 0x0" ::: "memory");   // my wave's fills done
    }
    __syncthreads();                                        // all waves' fills done

    const int lane = threadIdx.x & 31;
    const int wave = threadIdx.x >> 5;
    const int half = lane >> 4;      // 0: K pair {0,1}; 1: K pair {2,3}
    const int lrow = lane & 15;      // A row (M), B/C column (N) within tile
    const int row_base = (blockIdx.x * 8 + wave) * 16;

    int arow = row_base + lrow;
    if (arow >= NN) arow = NN - 1;              // clamp: keep EXEC all-ones for WMMA
    const float* xrow = x + (size_t)arow * IN_CH;

    v8f acc[8] = {};                            // 8 x (16x16 f32) accumulators

    for (int k0 = 0; k0 < IN_CH; k0 += 4) {
        const int ka = k0 + 2 * half;           // this half-wave's K pair
        const v2f a = *(const v2f*)(xrow + ka); // A: comp0=K=ka, comp1=K=ka+1
        const float* swk = &sw[(size_t)ka * HD];
#pragma unroll
        for (int nt = 0; nt < 8; ++nt) {
            const int n = nt * 16 + lrow;
            v2f b;
            b.x = swk[n];                       // B: comp0=K=ka
            b.y = swk[HD + n];                  //    comp1=K=ka+1
            acc[nt] = __builtin_amdgcn_wmma_f32_16x16x4_f32(
                false, a, false, b, (short)0, acc[nt], false, false);
        }
    }

    // C/D layout: VGPR v, lanes 0-15 -> M=v, lanes 16-31 -> M=v+8; N=lrow.
#pragma unroll
    for (int nt = 0; nt < 8; ++nt) {
#pragma unroll
        for (int v = 0; v < 8; ++v) {
            const int row = row_base + half * 8 + v;
            if (row < NN) h[(size_t)row * HD + nt * 16 + lrow] = acc[nt][v];
        }
    }
}

// ---------------------------------------------------------------------------
// Kernel 2: per (node, head) attention logits  alpha_src / alpha_dst
// ---------------------------------------------------------------------------
__global__ void gat_alpha(const float* __restrict__ h, const float* __restrict__ att,
                          float* __restrict__ asrc, float* __restrict__ adst)
{
    const int t = blockIdx.x * blockDim.x + threadIdx.x;
    if (t >= NN * HEADS) return;
    const int node = t >> 2, hh = t & 3;
    const float4* hp = (const float4*)(h + (size_t)node * HD + hh * OUT_CH);
    const float4* as = (const float4*)(att + hh * 2 * OUT_CH);
    const float4* ad = (const float4*)(att + hh * 2 * OUT_CH + OUT_CH);
    float s = 0.f, d = 0.f;
#pragma unroll
    for (int i = 0; i < 8; ++i) {
        const float4 hv = hp[i], a1 = as[i], a2 = ad[i];
        s += hv.x * a1.x + hv.y * a1.y + hv.z * a1.z + hv.w * a1.w;
        d += hv.x * a2.x + hv.y * a2.y + hv.z * a2.z + hv.w * a2.w;
    }
    asrc[t] = s; adst[t] = d;
}

// ---------------------------------------------------------------------------
// Kernel 3: init emax=-inf (mono-coded), denom=0, out=bias broadcast
// ---------------------------------------------------------------------------
__global__ void gat_init(unsigned* __restrict__ emax, float* __restrict__ denom,
                         float* __restrict__ out, const float* __restrict__ bias)
{
    const int t = blockIdx.x * blockDim.x + threadIdx.x;
    if (t < NN * HEADS) { emax[t] = f2mono(-INFINITY); denom[t] = 0.f; }
    if (t < NN * HD)    { out[t] = bias[t & (HD - 1)]; }
}

__device__ __forceinline__ float lrelu(float e) {
    return e >= 0.f ? e : NEG_SLOPE * e;
}

// ---------------------------------------------------------------------------
// Kernel 4: segment max over dst via monotonic uint atomicMax (exact)
// ---------------------------------------------------------------------------
__global__ void gat_edge_max(const int* __restrict__ ei,
                             const float* __restrict__ asrc, const float* __restrict__ adst,
                             unsigned* __restrict__ emax)
{
    const int e = blockIdx.x * blockDim.x + threadIdx.x;
    if (e >= NE) return;
    const int s = ei[e], d = ei[NE + e];
    const float4 sa = *(const float4*)(asrc + (size_t)s * 4);
    const float4 da = *(const float4*)(adst + (size_t)d * 4);
    const float ev[4] = { lrelu(sa.x + da.x), lrelu(sa.y + da.y),
                          lrelu(sa.z + da.z), lrelu(sa.w + da.w) };
#pragma unroll
    for (int hh = 0; hh < 4; ++hh)
        atomicMax(emax + (size_t)d * 4 + hh, f2mono(ev[hh]));
}

// ---------------------------------------------------------------------------
// Kernel 5: softmax denominators via f32 atomic add
// ---------------------------------------------------------------------------
__global__ void gat_edge_den(const int* __restrict__ ei,
                             const float* __restrict__ asrc, const float* __restrict__ adst,
                             const unsigned* __restrict__ emax, float* __restrict__ denom)
{
    const int e = blockIdx.x * blockDim.x + threadIdx.x;
    if (e >= NE) return;
    const int s = ei[e], d = ei[NE + e];
    const float4 sa = *(const float4*)(asrc + (size_t)s * 4);
    const float4 da = *(const float4*)(adst + (size_t)d * 4);
    const uint4  mu = *(const uint4*)(emax + (size_t)d * 4);
    const float ev[4] = { lrelu(sa.x + da.x), lrelu(sa.y + da.y),
                          lrelu(sa.z + da.z), lrelu(sa.w + da.w) };
    const float mm[4] = { mono2f(mu.x), mono2f(mu.y), mono2f(mu.z), mono2f(mu.w) };
#pragma unroll
    for (int hh = 0; hh < 4; ++hh)
        atomicAdd(denom + (size_t)d * 4 + hh, __expf(ev[hh] - mm[hh]));
}

// ---------------------------------------------------------------------------
// Kernel 6: weighted scatter. One wave32 per edge; lane owns channels
// {lane, lane+32, lane+64, lane+96} -> 4 coalesced global_atomic_add_f32.
// ---------------------------------------------------------------------------
__global__ __launch_bounds__(256) void gat_scatter(
    const int* __restrict__ ei, const float* __restrict__ h,
    const float* __restrict__ asrc, const float* __restrict__ adst,
    const unsigned* __restrict__ emax, const float* __restrict__ denom,
    float* __restrict__ out)
{
    const long long gid = (long long)blockIdx.x * blockDim.x + threadIdx.x;
    const int e = (int)(gid >> 5);
    const int lane = (int)(gid & 31);
    if (e >= NE) return;
    const int s = ei[e], d = ei[NE + e];
    const float4 sa = *(const float4*)(asrc + (size_t)s * 4);
    const float4 da = *(const float4*)(adst + (size_t)d * 4);
    const uint4  mu = *(const uint4*)(emax + (size_t)d * 4);
    const float4 dn = *(const float4*)(denom + (size_t)d * 4);

    float alpha[4];
    alpha[0] = __expf(lrelu(sa.x + da.x) - mono2f(mu.x)) / dn.x;
    alpha[1] = __expf(lrelu(sa.y + da.y) - mono2f(mu.y)) / dn.y;
    alpha[2] = __expf(lrelu(sa.z + da.z) - mono2f(mu.z)) / dn.z;
    alpha[3] = __expf(lrelu(sa.w + da.w) - mono2f(mu.w)) / dn.w;

    const float* hs = h + (size_t)s * HD;
    float* od = out + (size_t)d * HD;
#pragma unroll
    for (int hh = 0; hh < 4; ++hh) {
        const int c = hh * 32 + lane;
        atomicAdd(od + c, alpha[hh] * hs[c]);
    }
}

// ---------------------------------------------------------------------------
extern "C" void kernel_launch(void* const* d_in, const int* in_sizes, int n_in,
                              void* d_out, int out_size, void* d_ws, size_t ws_size,
                              hipStream_t stream) {
    const float* x    = (const float*)d_in[0];   // (NN, 128)
    const int*   ei   = (const int*)  d_in[1];   // (2, NE)
    const float* w    = (const float*)d_in[2];   // (128, 128)
    const float* att  = (const float*)d_in[3];   // (4, 64)
    const float* bias = (const float*)d_in[4];   // (128,)
    float* out = (float*)d_out;                  // (NN, 128)

    // workspace layout
    float*    h     = (float*)d_ws;                            // NN*128 f32 (25.6 MB)
    float*    asrc  = h + (size_t)NN * HD;                     // NN*4
    float*    adst  = asrc + (size_t)NN * HEADS;               // NN*4
    unsigned* emax  = (unsigned*)(adst + (size_t)NN * HEADS);  // NN*4
    float*    denom = (float*)(emax + (size_t)NN * HEADS);     // NN*4

    const int row_tiles = (NN + 15) / 16;          // 3125
    const int gemm_blocks = (row_tiles + 7) / 8;   // 391

    gat_init<<<(NN * HD + 255) / 256, 256, 0, stream>>>(emax, denom, out, bias);
    gat_gemm_wmma<<<gemm_blocks, 256, 0, stream>>>(x, w, h);
    gat_alpha<<<(NN * HEADS + 255) / 256, 256, 0, stream>>>(h, att, asrc, adst);
    gat_edge_max<<<(NE + 255) / 256, 256, 0, stream>>>(ei, asrc, adst, emax);
    gat_edge_den<<<(NE + 255) / 256, 256, 0, stream>>>(ei, asrc, adst, emax, denom);
    gat_scatter<<<(int)(((long long)NE * 32 + 255) / 256), 256, 0, stream>>>(
        ei, h, asrc, adst, emax, denom, out);
}